// _UnfusedExperts_88673894793693
// MI455X (gfx1250) — compile-verified
//
#include <hip/hip_runtime.h>
#include <math.h>

// MoE sizes (compile-time, match reference)
#define NUM_EXPERTS 8
#define D_MODEL     2048
#define D_FF        4096
#define TOP_K       2
#define NUM_TOKENS  4096

#define MAX_PAIRS_PER_EXPERT (NUM_TOKENS * TOP_K)   // worst case: all pairs one expert
#define MTILES_MAX           (MAX_PAIRS_PER_EXPERT / 16)

typedef __attribute__((ext_vector_type(2))) float v2f;
typedef __attribute__((ext_vector_type(8))) float v8f;

#if defined(__has_builtin)
#if __has_builtin(__builtin_amdgcn_global_load_async_to_lds_b128)
#define HAVE_ASYNC_LDS 1
#endif
#endif

#ifdef HAVE_ASYNC_LDS
// Matches the builtin's parameter types exactly (from compiler diagnostic):
// param0: int __attribute__((vector_size(16))) addrspace(1)*  (global source)
// param1: LDS destination pointer (addrspace(3))
typedef int v4i_vs __attribute__((vector_size(16)));
typedef __attribute__((address_space(1))) v4i_vs* async_gptr;
typedef __attribute__((address_space(3))) v4i_vs* async_lptr;
#endif

// ---------------------------------------------------------------------------
// Kernel 1: zero output accumulator + per-expert pair counters
// ---------------------------------------------------------------------------
__global__ void k_moe_zero(float* __restrict__ out, int* __restrict__ cnts) {
    const size_t n4 = (size_t)NUM_TOKENS * D_MODEL / 4;
    float4* o4 = (float4*)out;
    for (size_t i = blockIdx.x * blockDim.x + threadIdx.x; i < n4;
         i += (size_t)gridDim.x * blockDim.x) {
        o4[i] = make_float4(0.f, 0.f, 0.f, 0.f);
    }
    if (blockIdx.x == 0 && threadIdx.x < NUM_EXPERTS) cnts[threadIdx.x] = 0;
}

// ---------------------------------------------------------------------------
// Kernel 2: route — group (token, weight) pairs by expert
// ---------------------------------------------------------------------------
__global__ void k_moe_route(const int* __restrict__ topk_idx,
                            const float* __restrict__ topk_w,
                            int* __restrict__ cnts,
                            int* __restrict__ rows,
                            float* __restrict__ wgts) {
    int gid = blockIdx.x * blockDim.x + threadIdx.x;
    if (gid >= NUM_TOKENS * TOP_K) return;
    int t = gid >> 1;
    int e = topk_idx[gid];
    float w = topk_w[gid];
    int pos = atomicAdd(&cnts[e], 1);
    rows[(size_t)e * MAX_PAIRS_PER_EXPERT + pos] = t;
    wgts[(size_t)e * MAX_PAIRS_PER_EXPERT + pos] = w;
}

// ---------------------------------------------------------------------------
// Kernel 3: fused grouped SwiGLU expert GEMM, fp32 WMMA
//   grid = (MTILES_MAX, NUM_EXPERTS), block = 512 threads (16 waves, wave32)
// ---------------------------------------------------------------------------
__global__ __launch_bounds__(512)
void k_moe_gemm(const float* __restrict__ X,        // [T][D]
                const float* __restrict__ Wg,       // [E][D][F]
                const float* __restrict__ Wu,       // [E][D][F]
                const float* __restrict__ Wd,       // [E][F][D]
                const int*   __restrict__ cnts,
                const int*   __restrict__ rows,
                const float* __restrict__ wgts,
                float*       __restrict__ out) {    // [T][D], atomic accum
    __shared__ float Xs[16 * 512];      // 32 KB staged activations chunk
    __shared__ float GUs[16 * 256];     // 16 KB: 16 tiles (8 gate + 8 up), 16x16 each
    __shared__ float Hs[16 * 128];      //  8 KB: silu(g)*u tile, row-major
    __shared__ int   toks[16];
    __shared__ float wrow[16];

    const int e     = blockIdx.y;
    const int mtile = blockIdx.x;
    const int cnt   = cnts[e];
    if (mtile * 16 >= cnt) return;

    const int tid  = threadIdx.x;
    const int wave = tid >> 5;          // 0..15
    const int lane = tid & 31;
    const int lrow = lane & 15;         // row (A) or column (B/C) sub-index
    const int khalf = (lane >> 4) * 2;  // 0 for lanes 0-15, 2 for lanes 16-31

    if (tid < 16) {
        int g = mtile * 16 + tid;
        if (g < cnt) {
            toks[tid] = rows[(size_t)e * MAX_PAIRS_PER_EXPERT + g];
            wrow[tid] = wgts[(size_t)e * MAX_PAIRS_PER_EXPERT + g];
        } else {
            toks[tid] = 0;
            wrow[tid] = 0.f;
        }
    }

    const float* WgE = Wg + (size_t)e * D_MODEL * D_FF;
    const float* WuE = Wu + (size_t)e * D_MODEL * D_FF;
    const float* WdE = Wd + (size_t)e * D_FF * D_MODEL;

    const int mat = wave >> 3;          // 0 = gate, 1 = up (phase A role)
    const int sub = wave & 7;           // which 16-col subtile of the F-tile
    const float* Wab = (mat == 0) ? WgE : WuE;

    v8f outacc[8] = {};                 // 8 n-tiles of 16 cols: this wave's D slice

    for (int fi = 0; fi < D_FF / 128; ++fi) {
        const int f0 = fi * 128;

        // ---------------- Phase A: G/U = X * W{g,u} -------------------------
        v8f acc = {};
        for (int kc = 0; kc < 4; ++kc) {
            const int kbase = kc * 512;
            __syncthreads();
            {   // cooperative stage of X chunk [16][512] (gathered rows)
                int r = tid >> 5;
                int c = (tid & 31) * 16;
                const float* gsrc = X + (size_t)toks[r] * D_MODEL + kbase + c;
                float*       ldst = Xs + r * 512 + c;
#ifdef HAVE_ASYNC_LDS
                // CDNA5 async global->LDS DMA (no VGPR bounce, tracked by ASYNCcnt).
                // Per ISA, the imm IOFFSET is added to BOTH global and LDS addresses.
                __builtin_amdgcn_global_load_async_to_lds_b128(
                    (async_gptr)gsrc, (async_lptr)ldst,  0, 0);
                __builtin_amdgcn_global_load_async_to_lds_b128(
                    (async_gptr)gsrc, (async_lptr)ldst, 16, 0);
                __builtin_amdgcn_global_load_async_to_lds_b128(
                    (async_gptr)gsrc, (async_lptr)ldst, 32, 0);
                __builtin_amdgcn_global_load_async_to_lds_b128(
                    (async_gptr)gsrc, (async_lptr)ldst, 48, 0);
#if __has_builtin(__builtin_amdgcn_s_wait_asynccnt)
                __builtin_amdgcn_s_wait_asynccnt(0);
#else
                asm volatile("s_wait_asynccnt 0" ::: "memory");
#endif
#else
                const float4* src = (const float4*)gsrc;
                float4* dst = (float4*)ldst;
                dst[0] = src[0]; dst[1] = src[1]; dst[2] = src[2]; dst[3] = src[3];
#endif
            }
            __syncthreads();

            const float* wb = Wab + (size_t)(kbase + khalf) * D_FF + (f0 + sub * 16 + lrow);
            __builtin_prefetch(wb + (size_t)512 * D_FF, 0, 3);
            #pragma unroll 4
            for (int ks = 0; ks < 128; ++ks) {
                const int kk = ks * 4 + khalf;
                v2f a;
                const float* xp = Xs + lrow * 512 + kk;
                a.x = xp[0]; a.y = xp[1];
                v2f b;
                const float* bp = wb + (size_t)ks * 4 * D_FF;
                b.x = bp[0]; b.y = bp[D_FF];
                acc = __builtin_amdgcn_wmma_f32_16x16x4_f32(
                        false, a, false, b, (short)0, acc, false, false);
            }
        }
        // write this wave's 16x16 G or U tile to LDS
        #pragma unroll
        for (int r = 0; r < 8; ++r) {
            int mrow = r + 8 * (lane >> 4);
            GUs[wave * 256 + mrow * 16 + lrow] = acc[r];
        }
        __syncthreads();

        // ---------------- Elementwise: H = silu(G) * U ----------------------
        #pragma unroll
        for (int i = 0; i < 4; ++i) {
            int id  = tid + i * 512;      // 0..2047
            int row = id >> 7;            // 0..15
            int col = id & 127;           // 0..127
            int s2  = col >> 4;
            int ic  = col & 15;
            float g = GUs[s2 * 256 + row * 16 + ic];
            float u = GUs[(8 + s2) * 256 + row * 16 + ic];
            float sg = g / (1.f + expf(-g));
            Hs[row * 128 + col] = sg * u;
        }
        __syncthreads();

        // ---------------- Phase B: out += H * Wd ----------------------------
        const int dbase = wave * 128;
        const float* wdb = WdE + (size_t)(f0 + khalf) * D_MODEL + dbase + lrow;
        __builtin_prefetch(wdb + (size_t)128 * D_MODEL, 0, 3);
        for (int ks = 0; ks < 32; ++ks) {
            const int kk = ks * 4 + khalf;
            v2f a;
            const float* hp = Hs + lrow * 128 + kk;
            a.x = hp[0]; a.y = hp[1];
            const float* bp0 = wdb + (size_t)ks * 4 * D_MODEL;
            #pragma unroll
            for (int nt = 0; nt < 8; ++nt) {
                v2f b;
                const float* bp = bp0 + nt * 16;
                b.x = bp[0]; b.y = bp[D_MODEL];
                outacc[nt] = __builtin_amdgcn_wmma_f32_16x16x4_f32(
                        false, a, false, b, (short)0, outacc[nt], false, false);
            }
        }
        __syncthreads();   // protect Hs/GUs before next f-tile overwrites
    }

    // ---------------- scatter-accumulate weighted expert output -------------
    #pragma unroll
    for (int r = 0; r < 8; ++r) {
        int mrow = r + 8 * (lane >> 4);
        float w = wrow[mrow];
        if (w != 0.f) {
            int t = toks[mrow];
            float* orow = out + (size_t)t * D_MODEL + wave * 128 + lrow;
            #pragma unroll
            for (int nt = 0; nt < 8; ++nt) {
                atomicAdd(orow + nt * 16, outacc[nt][r] * w);
            }
        }
    }
}

// ---------------------------------------------------------------------------
// Host-side launcher
// ---------------------------------------------------------------------------
extern "C" void kernel_launch(void* const* d_in, const int* in_sizes, int n_in,
                              void* d_out, int out_size, void* d_ws, size_t ws_size,
                              hipStream_t stream) {
    const float* hidden = (const float*)d_in[0];
    const int*   tk_idx = (const int*)d_in[1];
    const float* tk_w   = (const float*)d_in[2];
    const float* Wg     = (const float*)d_in[3];
    const float* Wu     = (const float*)d_in[4];
    const float* Wd     = (const float*)d_in[5];
    float* out = (float*)d_out;

    // workspace layout
    int*   cnts = (int*)d_ws;                                   // 8 ints
    int*   rows = (int*)((char*)d_ws + 256);                    // E * 8192 ints
    float* wgts = (float*)((char*)d_ws + 256 +
                           sizeof(int) * NUM_EXPERTS * MAX_PAIRS_PER_EXPERT);

    k_moe_zero<<<2048, 256, 0, stream>>>(out, cnts);

    k_moe_route<<<(NUM_TOKENS * TOP_K + 255) / 256, 256, 0, stream>>>(
        tk_idx, tk_w, cnts, rows, wgts);

    dim3 grid(MTILES_MAX, NUM_EXPERTS);
    k_moe_gemm<<<grid, 512, 0, stream>>>(hidden, Wg, Wu, Wd, cnts, rows, wgts, out);
}